// DriverDenseModel_66907000537375
// MI455X (gfx1250) — compile-verified
//
#include <hip/hip_runtime.h>

// CDNA5 / gfx1250, wave32. Fused DLRM-style MLP:
//   h   = relu(dense @ W1[0:64,:] + sum_j W1[64+off_j+idx_j,:] + b1)
//   out = h @ W2 + b2
// All matrix math via V_WMMA_F32_16X16X4_F32 (exact f32 semantics, RNE).
// Weights (2.3 MB) are L2-resident (192 MB L2); h never touches HBM.

typedef float v2f __attribute__((ext_vector_type(2)));
typedef float v8f __attribute__((ext_vector_type(8)));

#define NDENSE   64
#define HID      256
#define OUT_DIM  256
#define MTILE    32          // batch rows per workgroup
#define DSTRIDE  68          // 64 + 4 pad: bank = (4*row + k) % 64 -> conflict-free
#define HSTRIDE  260         // 256 + 4 pad: same property

__global__ __launch_bounds__(256)
void fused_dlrm_mlp(const float* __restrict__ dense,
                    const int*   __restrict__ sparse,   // [B,8] indices
                    const float* __restrict__ W1,       // [1994,256] row-major
                    const float* __restrict__ b1,       // [256]
                    const float* __restrict__ W2,       // [256,256] row-major
                    const float* __restrict__ b2,       // [256]
                    float* __restrict__ out)            // [B,256]
{
    __shared__ float lds_dense[MTILE * DSTRIDE];
    __shared__ float lds_h[MTILE * HSTRIDE];     // holds e (=emb+b1), then h
    __shared__ int   lds_idx[MTILE * 8];

    const int tid     = threadIdx.x;
    const int lane    = tid & 31;
    const int wave    = tid >> 5;
    const int rowBase = blockIdx.x * MTILE;

    // Row offsets of each sparse feature's one-hot block inside W1.
    // prefix sums of CARDS = (1000,500,200,100,50,50,20,10), +64 dense rows.
    const int off[8] = {64 + 0, 64 + 1000, 64 + 1500, 64 + 1700,
                        64 + 1800, 64 + 1850, 64 + 1900, 64 + 1920};

    // Warm L2 tags for W2 (one cacheline per row start). gfx1250 global_prefetch.
    __builtin_prefetch(&W2[tid * OUT_DIM], 0, 1);

    // ---- Stage 1: cooperative loads into LDS ------------------------------
    // 32*8 = 256 indices: exactly one per thread.
    lds_idx[tid] = sparse[rowBase * 8 + tid];
    // 32x64 dense tile, 8 elements per thread, coalesced.
#pragma unroll
    for (int i = 0; i < (MTILE * NDENSE) / 256; ++i) {
        int flat = tid + i * 256;
        int r = flat >> 6, c = flat & 63;
        lds_dense[r * DSTRIDE + c] = dense[(size_t)(rowBase + r) * NDENSE + c];
    }
    __syncthreads();

    // ---- Stage 2: e = b1 + sum of 8 embedding rows (column-per-thread) ----
    {
        const int c = tid;            // 256 threads == 256 columns
        const float bias = b1[c];
#pragma unroll 4
        for (int r = 0; r < MTILE; ++r) {
            float s = bias;
#pragma unroll
            for (int j = 0; j < 8; ++j) {
                int row = off[j] + lds_idx[r * 8 + j];     // L2-resident gather
                s += W1[(size_t)row * HID + c];
            }
            lds_h[r * HSTRIDE + c] = s;
        }
    }
    __syncthreads();

    // ---- Wave tiling: 2 (M) x 4 (N) waves -> disjoint 16x64 strips --------
    const int m0    = (wave >> 2) * 16;       // 0 or 16
    const int n0    = (wave & 3)  * 64;       // 0,64,128,192
    const int nLane = lane & 15;              // N within 16-wide tile (B/C/D)
    const int aRow  = lane & 15;              // M within A tile
    const int kHi   = (lane >> 4) * 2;        // A/B: lanes 16-31 hold K+2
    const int mHi   = (lane >> 4) * 8;        // C/D: lanes 16-31 hold M+8

    // Per-lane hoisted base pointers (hot loops become imm-offset accesses).
    const float* aDense = &lds_dense[(m0 + aRow) * DSTRIDE + kHi];
    const float* aH     = &lds_h[(m0 + aRow) * HSTRIDE + kHi];
    float*       cH     = &lds_h[(m0 + mHi) * HSTRIDE + n0 + nLane];

    // ---- Phase 1: h = relu(dense @ W1_dense + e) --------------------------
    v8f acc[4];
#pragma unroll
    for (int t = 0; t < 4; ++t)
#pragma unroll
        for (int v = 0; v < 8; ++v)
            acc[t][v] = cH[v * HSTRIDE + t * 16];

    // Per-lane base into W1 dense block: row kHi, column (n0+nLane).
    const float* w1p = W1 + (size_t)kHi * HID + n0 + nLane;
#pragma unroll
    for (int k0 = 0; k0 < NDENSE; k0 += 4) {
        v2f a;   // A 16x4 f32: comp i -> K = k0 + kHi + i, M = aRow
        a.x = aDense[k0];
        a.y = aDense[k0 + 1];
#pragma unroll
        for (int t = 0; t < 4; ++t) {
            v2f b;   // B 4x16 f32: comp i -> K = k0 + kHi + i, N = n0+16t+nLane
            b.x = w1p[(size_t)k0 * HID + t * 16];
            b.y = w1p[(size_t)(k0 + 1) * HID + t * 16];
            acc[t] = __builtin_amdgcn_wmma_f32_16x16x4_f32(
                false, a, false, b, (short)0, acc[t], false, false);
        }
    }

    // ReLU in registers, write h into this wave's OWN LDS strip (no race).
#pragma unroll
    for (int t = 0; t < 4; ++t)
#pragma unroll
        for (int v = 0; v < 8; ++v) {
            float hv = acc[t][v];
            cH[v * HSTRIDE + t * 16] = hv > 0.0f ? hv : 0.0f;
        }
    __syncthreads();   // phase 2 reads all 256 columns of h

    // ---- Phase 2: out = h @ W2 + b2  (fully unrolled K=256) ---------------
#pragma unroll
    for (int t = 0; t < 4; ++t) {
        float bb = b2[n0 + t * 16 + nLane];
#pragma unroll
        for (int v = 0; v < 8; ++v) acc[t][v] = bb;
    }

    const float* w2p = W2 + (size_t)kHi * OUT_DIM + n0 + nLane;
#pragma unroll
    for (int k0 = 0; k0 < HID; k0 += 4) {
        v2f a;
        a.x = aH[k0];
        a.y = aH[k0 + 1];
#pragma unroll
        for (int t = 0; t < 4; ++t) {
            v2f b;
            b.x = w2p[(size_t)k0 * OUT_DIM + t * 16];
            b.y = w2p[(size_t)(k0 + 1) * OUT_DIM + t * 16];
            acc[t] = __builtin_amdgcn_wmma_f32_16x16x4_f32(
                false, a, false, b, (short)0, acc[t], false, false);
        }
    }

    // ---- Store out (lanes 0-15 write 64B contiguous, coalesced) -----------
    float* outp = out + (size_t)(rowBase + m0 + mHi) * OUT_DIM + n0 + nLane;
#pragma unroll
    for (int t = 0; t < 4; ++t)
#pragma unroll
        for (int v = 0; v < 8; ++v)
            outp[(size_t)v * OUT_DIM + t * 16] = acc[t][v];
}

extern "C" void kernel_launch(void* const* d_in, const int* in_sizes, int n_in,
                              void* d_out, int out_size, void* d_ws, size_t ws_size,
                              hipStream_t stream) {
    const float* dense  = (const float*)d_in[0];
    const int*   sparse = (const int*)  d_in[1];
    const float* W1     = (const float*)d_in[2];
    const float* b1     = (const float*)d_in[3];
    const float* W2     = (const float*)d_in[4];
    const float* b2     = (const float*)d_in[5];
    float* out = (float*)d_out;

    const int B = in_sizes[0] / NDENSE;          // 16384
    dim3 grid(B / MTILE);                        // 512 workgroups
    fused_dlrm_mlp<<<grid, 256, 0, stream>>>(dense, sparse, W1, b1, W2, b2, out);
}